// spss_pcbm_58342835748996
// MI455X (gfx1250) — compile-verified
//
#include <hip/hip_runtime.h>
#include <math.h>

typedef __attribute__((ext_vector_type(2))) float v2f;
typedef __attribute__((ext_vector_type(8))) float v8f;
typedef __attribute__((ext_vector_type(4))) unsigned int tdm_u32x4;
typedef __attribute__((ext_vector_type(8))) int tdm_i32x8;
typedef __attribute__((ext_vector_type(4))) int tdm_i32x4;

#define B_   64
#define N_   1024
#define TW_  1280
#define K_   512
#define NC_  1000
#define EMB_ 768

#if defined(__HIP_DEVICE_COMPILE__) && __has_builtin(__builtin_amdgcn_tensor_load_to_lds)
#define HAVE_TDM 1
#else
#define HAVE_TDM 0
#endif

// ---------------------------------------------------------------------------
// Kernel A: qk[c] = K^-0.5 * sum_d ( sum_k mean(cavs)[k]*wq_w[d,k] ) * wk_w[d,c]
// Collapses the entire key-projection GEMM into one 512-vector.
// ---------------------------------------------------------------------------
__global__ __launch_bounds__(512)
void qk_precompute(const float* __restrict__ cavs,
                   const float* __restrict__ wq_w,
                   const float* __restrict__ wk_w,
                   float* __restrict__ qk)
{
    __shared__ float s_pc[K_];
    __shared__ float s_q[K_];
    const int tid = threadIdx.x;   // 512 threads, one per channel

    {   // pooled_cavs[k] = mean over EMB
        const float* row = cavs + (size_t)tid * EMB_;
        float s = 0.f;
        for (int e = 0; e < EMB_; ++e) s += row[e];
        s_pc[tid] = s * (1.0f / EMB_);
    }
    __syncthreads();
    {   // q[d] = sum_k pc[k] * wq_w[d,k]
        const float* row = wq_w + (size_t)tid * K_;
        float s = 0.f;
        for (int k = 0; k < K_; ++k) s += s_pc[k] * row[k];
        s_q[tid] = s;
    }
    __syncthreads();
    {   // qk[c] = scale * sum_d q[d] * wk_w[d,c]   (coalesced over c)
        float s = 0.f;
        for (int d = 0; d < K_; ++d) s += s_q[d] * wk_w[(size_t)d * K_ + tid];
        qk[tid] = s * 0.04419417382415922f;   // 512^-0.5
    }
}

// ---------------------------------------------------------------------------
// Kernel B: token_concepts = patches @ proj_w^T + proj_b  via V_WMMA_F32_16X16X4_F32
// Workgroup: 256 threads (8 waves). Tile: 32 rows x 512 cols.
// Each wave: 32x64 sub-tile = 2x4 fragments of 16x16 f32.
// A tile staged by the Tensor Data Mover when available (pad 16+2 DWORDs ->
// the [32][18] anti-conflict layout), else manual staging.
// ---------------------------------------------------------------------------
__global__ __launch_bounds__(256)
void token_proj_wmma(const float* __restrict__ patches,
                     const float* __restrict__ proj_w,
                     const float* __restrict__ proj_b,
                     float* __restrict__ tc)
{
    __shared__ float  As[32][18];     // 32 rows x 16 k, padded to 18
    __shared__ float2 Bs[8][K_];      // k-pair p -> (t=2p, 2p+1), 512 cols

    const int tid  = threadIdx.x;
    const int lane = tid & 31;
    const int wave = tid >> 5;
    const int l16  = lane & 15;
    const int hi   = lane >> 4;       // 0: lanes 0-15, 1: lanes 16-31
    const int row0 = blockIdx.x * 32;
    const int cb   = wave * 64;       // this wave's 64-column block

    v8f acc[2][4];
    #pragma unroll
    for (int m = 0; m < 2; ++m)
        #pragma unroll
        for (int n = 0; n < 4; ++n)
            #pragma unroll
            for (int r = 0; r < 8; ++r) acc[m][n][r] = 0.0f;

#if HAVE_TDM
    // LDS byte offset of As (low 32 bits of the flat shared address)
    const unsigned lds_as = (unsigned)(unsigned long long)(const void*)&As[0][0];
#endif

    for (int t0 = 0; t0 < TW_; t0 += 16) {
        __syncthreads();   // previous chunk fully consumed before overwrite

#if HAVE_TDM
        if (wave == 0) {
            // D# for a 2D tile: 16 (dim0) x 32 rows (dim1), f32, stride 1280,
            // LDS padding: 2 DWORDs every 16 DWORDs -> rows of 18 floats.
            const unsigned long long ga =
                (unsigned long long)(const void*)(patches + (size_t)row0 * TW_ + t0);
            tdm_u32x4 g0;
            g0[0] = 1u;                                     // count=1, user mode
            g0[1] = lds_as;                                 // lds_addr
            g0[2] = (unsigned)ga;                           // global_addr[31:0]
            g0[3] = (unsigned)((ga >> 32) & 0x01FFFFFFull)  // global_addr[56:32]
                  | (2u << 30);                             // type=2 (image)
            tdm_i32x8 g1;
            g1[0] = (2 << 16)      // data_size = 4B
                  | (1 << 20)      // pad_enable
                  | (3 << 22)      // pad_interval: 16 DWORDs
                  | (1 << 25);     // pad_amount: 2 DWORDs
            g1[1] = (int)((TW_ & 0xFFFF) << 16);            // tensor_dim0 lo
            g1[2] = (int)(TW_ >> 16);                       // dim0 hi | dim1 lo(=0)
            g1[3] = (int)(((B_ * N_) >> 16) | (16 << 16));  // dim1 hi | tile_dim0=16
            g1[4] = 32;                                     // tile_dim1=32, tile_dim2=0
            g1[5] = TW_;                                    // tensor_dim0_stride lo
            g1[6] = 0;
            g1[7] = 0;
            tdm_i32x4 gz = {0, 0, 0, 0};
#if defined(__clang_major__) && __clang_major__ >= 23
            tdm_i32x8 gz8 = {0, 0, 0, 0, 0, 0, 0, 0};
            __builtin_amdgcn_tensor_load_to_lds(g0, g1, gz, gz, gz8, 0);
#else
            __builtin_amdgcn_tensor_load_to_lds(g0, g1, gz, gz, 0);
#endif
        }
#else
        for (int e = tid; e < 512; e += 256) {
            const int r = e >> 4, t = e & 15;
            As[r][t] = patches[(size_t)(row0 + r) * TW_ + t0 + t];
        }
#endif

        // stage B transposed into k-pairs: Bs[p][c] = proj_w[c][t0+2p .. +2p+1]
        for (int c = tid; c < K_; c += 256) {
            const float* src = proj_w + (size_t)c * TW_ + t0;
            float4 v0 = *(const float4*)(src + 0);
            float4 v1 = *(const float4*)(src + 4);
            float4 v2 = *(const float4*)(src + 8);
            float4 v3 = *(const float4*)(src + 12);
            Bs[0][c] = make_float2(v0.x, v0.y);
            Bs[1][c] = make_float2(v0.z, v0.w);
            Bs[2][c] = make_float2(v1.x, v1.y);
            Bs[3][c] = make_float2(v1.z, v1.w);
            Bs[4][c] = make_float2(v2.x, v2.y);
            Bs[5][c] = make_float2(v2.z, v2.w);
            Bs[6][c] = make_float2(v3.x, v3.y);
            Bs[7][c] = make_float2(v3.z, v3.w);
        }

#if HAVE_TDM
#if __has_builtin(__builtin_amdgcn_s_wait_tensorcnt)
        if (wave == 0) __builtin_amdgcn_s_wait_tensorcnt(0);
#endif
#endif
        __syncthreads();

        #pragma unroll
        for (int kk = 0; kk < 4; ++kk) {
            const int tA = kk * 4 + hi * 2;   // this lane's k-pair start
            // A fragments (16x4 f32): lane<16 -> K={0,1}, lane>=16 -> K={2,3}
            v2f a0, a1;
            a0[0] = As[l16][tA];       a0[1] = As[l16][tA + 1];
            a1[0] = As[16 + l16][tA];  a1[1] = As[16 + l16][tA + 1];
            // B fragments: one aligned b64 read per tile
            const int p = kk * 2 + hi;
            v2f bf[4];
            #pragma unroll
            for (int n = 0; n < 4; ++n) {
                const float2 t2 = Bs[p][cb + n * 16 + l16];
                bf[n][0] = t2.x;
                bf[n][1] = t2.y;
            }
            #pragma unroll
            for (int n = 0; n < 4; ++n) {
                acc[0][n] = __builtin_amdgcn_wmma_f32_16x16x4_f32(
                    false, a0, false, bf[n], (short)0, acc[0][n], false, false);
                acc[1][n] = __builtin_amdgcn_wmma_f32_16x16x4_f32(
                    false, a1, false, bf[n], (short)0, acc[1][n], false, false);
            }
        }
    }

    // epilogue: + proj_b, store token_concepts
    // C/D layout: VGPR r, lanes 0-15 -> M=r ; lanes 16-31 -> M=r+8 ; N = lane%16
    #pragma unroll
    for (int n = 0; n < 4; ++n) {
        const int col = cb + n * 16 + l16;
        const float pb = proj_b[col];
        #pragma unroll
        for (int m = 0; m < 2; ++m) {
            const int rbase = row0 + m * 16 + hi * 8;
            #pragma unroll
            for (int r = 0; r < 8; ++r) {
                tc[(size_t)(rbase + r) * K_ + col] = acc[m][n][r] + pb;
            }
        }
    }
}

// ---------------------------------------------------------------------------
// Kernel C: one block per batch. Per-token LN + attn logit -> softmax ->
// attention pooling -> classifier LN -> classifier matvec.
// ---------------------------------------------------------------------------
__global__ __launch_bounds__(256)
void pool_classify(const float* __restrict__ tc,
                   const float* __restrict__ qk,
                   const float* __restrict__ ln_w,
                   const float* __restrict__ ln_b,
                   const float* __restrict__ cls_ln_w,
                   const float* __restrict__ cls_ln_b,
                   const float* __restrict__ cls_w,
                   const float* __restrict__ cls_b,
                   float* __restrict__ out_logits,
                   float* __restrict__ out_pooled)
{
    __shared__ float s_attn[N_];
    __shared__ float s_mu[N_];
    __shared__ float s_rs[N_];
    __shared__ float s_red[256];
    __shared__ __align__(16) float s_vec[K_];
    __shared__ float s_stat[2];

    const int b    = blockIdx.x;
    const int tid  = threadIdx.x;
    const int lane = tid & 31;
    const int wave = tid >> 5;
    const float* tcb = tc + (size_t)b * N_ * K_;

    // phase 1: per-token LayerNorm stats (eps 1e-6) + logit = kv . qk
    for (int n = wave; n < N_; n += 8) {
        const float4* row4 = (const float4*)(tcb + (size_t)n * K_);
        float4 vals[4];
        float s = 0.f, sq = 0.f;
        #pragma unroll
        for (int i = 0; i < 4; ++i) {
            const float4 v = row4[lane + 32 * i];
            vals[i] = v;
            s  += v.x + v.y + v.z + v.w;
            sq += v.x * v.x + v.y * v.y + v.z * v.z + v.w * v.w;
        }
        #pragma unroll
        for (int off = 16; off >= 1; off >>= 1) {
            s  += __shfl_xor(s,  off, 32);
            sq += __shfl_xor(sq, off, 32);
        }
        const float mu  = s * (1.0f / K_);
        const float var = sq * (1.0f / K_) - mu * mu;
        const float rs  = rsqrtf(var + 1e-6f);
        float dot = 0.f;
        #pragma unroll
        for (int i = 0; i < 4; ++i) {
            const int g = lane + 32 * i;
            const float4 w4 = ((const float4*)ln_w)[g];
            const float4 b4 = ((const float4*)ln_b)[g];
            const float4 q4 = ((const float4*)qk)[g];
            const float4 v  = vals[i];
            dot += ((v.x - mu) * rs * w4.x + b4.x) * q4.x;
            dot += ((v.y - mu) * rs * w4.y + b4.y) * q4.y;
            dot += ((v.z - mu) * rs * w4.z + b4.z) * q4.z;
            dot += ((v.w - mu) * rs * w4.w + b4.w) * q4.w;
        }
        #pragma unroll
        for (int off = 16; off >= 1; off >>= 1) dot += __shfl_xor(dot, off, 32);
        if (lane == 0) { s_attn[n] = dot; s_mu[n] = mu; s_rs[n] = rs; }
    }
    __syncthreads();

    // phase 2: softmax over N_=1024 logits
    {
        float m = -3.4e38f;
        for (int i = tid; i < N_; i += 256) m = fmaxf(m, s_attn[i]);
        s_red[tid] = m; __syncthreads();
        for (int off = 128; off >= 1; off >>= 1) {
            if (tid < off) s_red[tid] = fmaxf(s_red[tid], s_red[tid + off]);
            __syncthreads();
        }
        const float M = s_red[0];
        __syncthreads();
        float s = 0.f;
        for (int i = tid; i < N_; i += 256) {
            const float e = __expf(s_attn[i] - M);
            s_attn[i] = e; s += e;
        }
        s_red[tid] = s; __syncthreads();
        for (int off = 128; off >= 1; off >>= 1) {
            if (tid < off) s_red[tid] += s_red[tid + off];
            __syncthreads();
        }
        const float inv = 1.0f / s_red[0];
        __syncthreads();
        for (int i = tid; i < N_; i += 256) s_attn[i] *= inv;
        __syncthreads();
    }

    // phase 3: pooled[k] = sum_n attn[n] * kv[n,k]  (coalesced float2 reread)
    const int k0 = tid * 2;
    const float2 lw = ((const float2*)ln_w)[tid];
    const float2 lb = ((const float2*)ln_b)[tid];
    float px = 0.f, py = 0.f;
    for (int n = 0; n < N_; ++n) {
        const float a = s_attn[n], mu = s_mu[n], rs = s_rs[n];
        const float2 r = ((const float2*)(tcb + (size_t)n * K_))[tid];
        px += a * ((r.x - mu) * rs * lw.x + lb.x);
        py += a * ((r.y - mu) * rs * lw.y + lb.y);
    }
    ((float2*)(out_pooled + (size_t)b * K_))[tid] = make_float2(px, py);

    // phase 4: classifier LayerNorm (eps 1e-5)
    s_red[tid] = px + py; __syncthreads();
    for (int off = 128; off >= 1; off >>= 1) {
        if (tid < off) s_red[tid] += s_red[tid + off];
        __syncthreads();
    }
    if (tid == 0) s_stat[0] = s_red[0] * (1.0f / K_);
    __syncthreads();
    s_red[tid] = px * px + py * py; __syncthreads();
    for (int off = 128; off >= 1; off >>= 1) {
        if (tid < off) s_red[tid] += s_red[tid + off];
        __syncthreads();
    }
    if (tid == 0) {
        const float mu  = s_stat[0];
        const float var = s_red[0] * (1.0f / K_) - mu * mu;
        s_stat[1] = rsqrtf(var + 1e-5f);
    }
    __syncthreads();
    {
        const float mu = s_stat[0], rs = s_stat[1];
        s_vec[k0]     = (px - mu) * rs * cls_ln_w[k0]     + cls_ln_b[k0];
        s_vec[k0 + 1] = (py - mu) * rs * cls_ln_w[k0 + 1] + cls_ln_b[k0 + 1];
    }
    __syncthreads();

    // phase 5: logits = h @ cls_w^T + cls_b   (wave per output row, coalesced)
    for (int j = wave; j < NC_; j += 8) {
        const float4* wrow = (const float4*)(cls_w + (size_t)j * K_);
        float acc = 0.f;
        #pragma unroll
        for (int i = 0; i < 4; ++i) {
            const int g = lane + 32 * i;
            const float4 w4 = wrow[g];
            const float4 h4 = ((const float4*)s_vec)[g];
            acc += w4.x * h4.x + w4.y * h4.y + w4.z * h4.z + w4.w * h4.w;
        }
        #pragma unroll
        for (int off = 16; off >= 1; off >>= 1) acc += __shfl_xor(acc, off, 32);
        if (lane == 0) out_logits[(size_t)b * NC_ + j] = cls_b[j] + acc;
    }
}

// ---------------------------------------------------------------------------
extern "C" void kernel_launch(void* const* d_in, const int* in_sizes, int n_in,
                              void* d_out, int out_size, void* d_ws, size_t ws_size,
                              hipStream_t stream) {
    const float* patches  = (const float*)d_in[0];
    const float* cavs     = (const float*)d_in[1];
    const float* proj_w   = (const float*)d_in[2];
    const float* proj_b   = (const float*)d_in[3];
    const float* ln_w     = (const float*)d_in[4];
    const float* ln_b     = (const float*)d_in[5];
    const float* wq_w     = (const float*)d_in[6];
    const float* wk_w     = (const float*)d_in[7];
    const float* cls_ln_w = (const float*)d_in[8];
    const float* cls_ln_b = (const float*)d_in[9];
    const float* cls_w    = (const float*)d_in[10];
    const float* cls_b    = (const float*)d_in[11];

    float* out        = (float*)d_out;
    float* out_logits = out;                          // [64,1000]
    float* out_pooled = out + (size_t)B_ * NC_;       // [64,512]
    float* out_tc     = out_pooled + (size_t)B_ * K_; // [64,1024,512]

    float* qk = (float*)d_ws;                         // 512 floats scratch

    qk_precompute<<<1, 512, 0, stream>>>(cavs, wq_w, wk_w, qk);
    token_proj_wmma<<<(B_ * N_) / 32, 256, 0, stream>>>(patches, proj_w, proj_b, out_tc);
    pool_classify<<<B_, 256, 0, stream>>>(out_tc, qk, ln_w, ln_b,
                                          cls_ln_w, cls_ln_b, cls_w, cls_b,
                                          out_logits, out_pooled);
}